// CRPSloss_53489522705022
// MI455X (gfx1250) — compile-verified
//
#include <hip/hip_runtime.h>
#include <stdint.h>

// Problem constants from the reference (B=8, N=16, C=4, H=128, W=256).
#define P_TOTAL     1048576u          // B*C*H*W pixels
#define CHW         131072u           // C*H*W  (stride between ensemble members)
#define NENS        16
#define TILE        256u              // pixels per tile (one block iteration)
#define NTILES      (P_TOTAL / TILE)  // 4096
#define TILES_PER_B (CHW / TILE)      // 512
#define NBLOCKS     1024
#define NTHREADS    256

typedef __attribute__((ext_vector_type(2))) float v2f;
typedef __attribute__((ext_vector_type(8))) float v8f;

// Low 32 bits of a generic pointer to LDS == workgroup-relative LDS byte
// address (flat aperture layout; async-to-LDS VDST wants exactly this).
__device__ __forceinline__ uint32_t lds_addr_u32(const void* p) {
    return (uint32_t)(uintptr_t)p;
}

// CDNA5 async copy: 16B per enabled lane, global -> LDS, tracked by ASYNCcnt.
__device__ __forceinline__ void async_load_b128(uint32_t lds_addr, const float* gaddr) {
    asm volatile("global_load_async_to_lds_b128 %0, %1, off"
                 :: "v"(lds_addr), "v"(gaddr)
                 : "memory");
}

__device__ __forceinline__ void wait_async0() {
#if defined(__has_builtin) && __has_builtin(__builtin_amdgcn_s_wait_asynccnt)
    __builtin_amdgcn_s_wait_asynccnt(0);
#else
    asm volatile("s_wait_asynccnt 0x0" ::: "memory");
#endif
}

// Exact 32-lane sum via the matrix unit. Correct independent of the B-matrix
// lane mapping because B is all-ones: D(m,n) = sum_k A(m,k) + C(m,n).
// Documented f32 A layout: lanes 0-15 supply K=0/1, lanes 16-31 supply K=2/3,
// so with a={acc,0}: RowSum(m) = acc(m) + acc(m+16) = D(m,*).
// Documented D layout: lanes 0-15 hold rows 0-7 in d[0..7], lanes 16-31 hold
// rows 8-15 -> per-lane sum of d[0..7] is a half-sum; shfl_xor(16) completes.
// Requires EXEC all-ones (call only from uniform control flow, full waves).
__device__ __forceinline__ float wave_reduce_wmma(float v) {
    v2f a; a[0] = v;    a[1] = 0.0f;
    v2f b; b[0] = 1.0f; b[1] = 1.0f;
    v8f c = {};
    v8f d = __builtin_amdgcn_wmma_f32_16x16x4_f32(
        /*neg_a=*/false, a, /*neg_b=*/false, b,
        /*c_mod=*/(short)0, c, /*reuse_a=*/false, /*reuse_b=*/false);
    float s = ((d[0] + d[1]) + (d[2] + d[3])) + ((d[4] + d[5]) + (d[6] + d[7]));
    s += __shfl_xor(s, 16, 32);
    return s;   // full 32-lane sum, in every lane
}

__global__ __launch_bounds__(NTHREADS) void crps_partial(
        const float* __restrict__ preds,   // [B, N, C, H, W]
        const float* __restrict__ gt,      // [B, C, H, W]
        float* __restrict__ partial) {     // [NBLOCKS]
    __shared__ __align__(16) float s_pred[NENS * TILE];  // 16 KB
    __shared__ __align__(16) float s_gt[TILE];           // 1 KB
    __shared__ float s_wred[NTHREADS / 32];

    const uint32_t t    = threadIdx.x;
    const uint32_t lane = t & 31u;
    const uint32_t wid  = t >> 5;

    const uint32_t lds_pred = lds_addr_u32(&s_pred[0]);
    const uint32_t lds_gt   = lds_addr_u32(&s_gt[0]);

    float acc = 0.0f;

    for (uint32_t tileIdx = blockIdx.x; tileIdx < NTILES; tileIdx += gridDim.x) {
        const uint32_t b      = tileIdx / TILES_PER_B;
        const uint32_t inner0 = (tileIdx - b * TILES_PER_B) * TILE;
        const float* gbase = gt    + (size_t)b * CHW + inner0;
        const float* pbase = preds + (size_t)b * ((size_t)NENS * CHW) + inner0;

        // Stage 16x256 pred slab: 1024 x 16B chunks, 4 per thread.
        #pragma unroll
        for (uint32_t k = 0; k < 4; ++k) {
            const uint32_t c   = t + k * NTHREADS;
            const uint32_t n   = c >> 6;            // 64 chunks per ensemble row
            const uint32_t col = (c & 63u) << 2;    // float offset within row
            async_load_b128(lds_pred + c * 16u, pbase + (size_t)n * CHW + col);
        }
        // Stage gt tile: 64 x 16B chunks; lanes 0..7 of every wave (uniform
        // ASYNCcnt increment per wave, EXEC nonzero per wave).
        if (lane < 8u) {
            const uint32_t c2 = wid * 8u + lane;
            async_load_b128(lds_gt + c2 * 16u, gbase + c2 * 4u);
        }

        wait_async0();        // this wave's async copies landed in LDS
        __syncthreads();      // all waves' copies landed

        const float g = s_gt[t];
        float x[NENS];
        #pragma unroll
        for (int n = 0; n < NENS; ++n) x[n] = s_pred[n * (int)TILE + (int)t];

        float t1 = 0.0f;
        #pragma unroll
        for (int i = 0; i < NENS; ++i) t1 += fabsf(x[i] - g);

        float t2 = 0.0f;
        #pragma unroll
        for (int i = 0; i < NENS; ++i)
            #pragma unroll
            for (int j = i + 1; j < NENS; ++j) t2 += fabsf(x[i] - x[j]);

        // term1 mean over N; term2: i<j sum / (N*(N-1))
        acc += t1 * (1.0f / NENS) - t2 * (1.0f / (NENS * (NENS - 1)));

        __syncthreads();      // protect LDS before next tile's async writes
    }

    // Deterministic block reduction: WMMA wave reduction, then across waves.
    const float s = wave_reduce_wmma(acc);
    if (lane == 0) s_wred[wid] = s;
    __syncthreads();
    if (t == 0) {
        float tot = 0.0f;
        #pragma unroll
        for (int i = 0; i < NTHREADS / 32; ++i) tot += s_wred[i];
        partial[blockIdx.x] = tot;
    }
}

__global__ __launch_bounds__(256) void crps_finalize(
        const float* __restrict__ partial, float* __restrict__ out) {
    __shared__ float s_wred[8];
    const uint32_t t = threadIdx.x, lane = t & 31u, wid = t >> 5;
    float acc = 0.0f;
    for (uint32_t i = t; i < NBLOCKS; i += 256u) acc += partial[i];
    const float s = wave_reduce_wmma(acc);
    if (lane == 0) s_wred[wid] = s;
    __syncthreads();
    if (t == 0) {
        float tot = 0.0f;
        #pragma unroll
        for (int i = 0; i < 8; ++i) tot += s_wred[i];
        out[0] = tot * (1.0f / (float)P_TOTAL);
    }
}

extern "C" void kernel_launch(void* const* d_in, const int* in_sizes, int n_in,
                              void* d_out, int out_size, void* d_ws, size_t ws_size,
                              hipStream_t stream) {
    (void)in_sizes; (void)n_in; (void)out_size; (void)ws_size;
    const float* preds = (const float*)d_in[0];
    const float* gt    = (const float*)d_in[1];
    float* partial     = (float*)d_ws;      // NBLOCKS floats = 4 KB scratch
    float* out         = (float*)d_out;

    crps_partial<<<NBLOCKS, NTHREADS, 0, stream>>>(preds, gt, partial);
    crps_finalize<<<1, 256, 0, stream>>>(partial, out);
}